// FiLM_28243704938789
// MI455X (gfx1250) — compile-verified
//
#include <hip/hip_runtime.h>
#include <math.h>

// FiLM forward for MI455X (gfx1250), wave32, WMMA f32 16x16x4.
//
//  k_gauss : per-scale bilinear discretization (f64 Gauss-Jordan w/ pivoting) -> AdT, Bd
//  k_em    : Legendre eval matrix (f64 recurrence) -> em[s][192][256]
//  k_bas   : DFT basis {cos, -sin} -> Bas[s][T][64]
//  k_weff  : fold irfft@t=191 phase into spectral weights -> Weff[s][16384][256]
//  k_scan  : HiPPO recurrence; AdT pre-swizzled into per-lane WMMA B-fragments
//            in LDS (256KB, ds_load_b64 per operand); state kept in LDS in
//            A-fragment layout; 16x256x256 per step via v_wmma_f32_16x16x4_f32
//  k_dft   : XF = cs-contraction with Bas   (GEMM, WMMA)
//  k_spec  : xout = XF @ Weff  (K=16384 GEMM, WMMA, 4 acc chains)
//  k_final : out = sum_s mlp_w[s] * (xout_s @ em_s^T) + mlp_b, transposed store

typedef __attribute__((ext_vector_type(2))) float v2f;
typedef __attribute__((ext_vector_type(4))) float v4f;
typedef __attribute__((ext_vector_type(8))) float v8f;

#define HID 256
#define NCH 512      // BATCH*NUM_FEAT = 8*64
#define OUTLEN 192
#define INLEN 768
#define TSLICE (NCH*HID)        // floats per time slice of cs (131072)
#define TWO_PI 6.283185307179586f

__device__ __forceinline__ v8f wmma4(v2f a, v2f b, v8f c) {
  // D = A(16x4,f32) * B(4x16,f32) + C(16x16,f32)
  return __builtin_amdgcn_wmma_f32_16x16x4_f32(false, a, false, b, (short)0, c,
                                               false, false);
}

// prefix of T over scales: 0, 192, 576  (T_s = 192<<s)
__device__ __forceinline__ int tprefix(int s) { return 192 * ((1 << s) - 1); }

// ---------------------------------------------------------------------------
// Constant builders
// ---------------------------------------------------------------------------

__global__ __launch_bounds__(256) void k_gauss(double* __restrict__ Gall,
                                               float* __restrict__ AdT,
                                               float* __restrict__ Bd) {
  const int s = blockIdx.x;
  const int tid = threadIdx.x;
  double* G = Gall + (size_t)s * 256 * 513;
  const double dt = 1.0 / (double)(192 << s);

  for (int idx = tid; idx < 256 * 513; idx += 256) {
    int r = idx / 513, c = idx - r * 513;
    double v;
    if (c < 512) {
      int j = c & 255;
      double sgn = (r < j) ? -1.0 : ((((r - j) & 1) == 0) ? -1.0 : 1.0);
      double A = sgn * (2.0 * r + 1.0);
      double eye = (r == j) ? 1.0 : 0.0;
      v = (c < 256) ? (eye - 0.5 * dt * A) : (eye + 0.5 * dt * A);
    } else {
      double Bc = ((r & 1) ? -1.0 : 1.0) * (2.0 * r + 1.0);
      v = dt * Bc;
    }
    G[idx] = v;
  }
  __syncthreads();

  __shared__ double sVal[256];
  __shared__ int sIdx[256];
  __shared__ int sPiv;

  for (int p = 0; p < 256; ++p) {
    sVal[tid] = (tid >= p) ? fabs(G[(size_t)tid * 513 + p]) : -1.0;
    sIdx[tid] = tid;
    __syncthreads();
    for (int st = 128; st > 0; st >>= 1) {
      if (tid < st && sVal[tid + st] > sVal[tid]) {
        sVal[tid] = sVal[tid + st];
        sIdx[tid] = sIdx[tid + st];
      }
      __syncthreads();
    }
    if (tid == 0) sPiv = sIdx[0];
    __syncthreads();
    int pr = sPiv;
    if (pr != p) {
      for (int c = tid; c < 513; c += 256) {
        double a = G[(size_t)p * 513 + c];
        G[(size_t)p * 513 + c] = G[(size_t)pr * 513 + c];
        G[(size_t)pr * 513 + c] = a;
      }
    }
    __syncthreads();
    double piv = G[(size_t)p * 513 + p];
    __syncthreads();
    double inv = 1.0 / piv;
    for (int c = tid; c < 513; c += 256) G[(size_t)p * 513 + c] *= inv;
    __syncthreads();
    int r = tid;
    if (r != p) {
      double f = G[(size_t)r * 513 + p];
      if (f != 0.0)
        for (int c = p; c < 513; ++c)
          G[(size_t)r * 513 + c] -= f * G[(size_t)p * 513 + c];
    }
    __syncthreads();
  }

  // AdT[k][n] = Ad[n][k] = G[n][256+k];  Bd[n] = G[n][512]
  for (int idx = tid; idx < 65536; idx += 256) {
    int k = idx >> 8, n = idx & 255;
    AdT[(size_t)s * 65536 + (size_t)k * 256 + n] =
        (float)G[(size_t)n * 513 + 256 + k];
  }
  Bd[s * 256 + tid] = (float)G[(size_t)tid * 513 + 512];
}

// 3 blocks x 192 threads: em[s][tt][n] = P_n(1 - 2*(T-192+tt)/T)
__global__ void k_em(float* __restrict__ em) {
  const int s = blockIdx.x;
  const int tt = threadIdx.x;  // 0..191
  const int T = 192 << s;
  const double x = 1.0 - 2.0 * (double)(T - 192 + tt) / (double)T;
  float* row = em + ((size_t)s * OUTLEN + tt) * 256;
  double p0 = 1.0, p1 = x;
  row[0] = 1.0f;
  row[1] = (float)x;
  for (int n = 1; n < 255; ++n) {
    double p2 = ((2.0 * n + 1.0) * x * p1 - (double)n * p0) / (double)(n + 1);
    row[n + 1] = (float)p2;
    p0 = p1;
    p1 = p2;
  }
}

// Bas[(tprefix(s)+t)*64 + kk] = kk<32 ? cos(2*pi*kk*t/T) : -sin(2*pi*(kk-32)*t/T)
__global__ void k_bas(float* __restrict__ Bas) {
  int gid = blockIdx.x * 256 + threadIdx.x;  // < 1344*64
  int tg = gid >> 6, kk = gid & 63;
  int s, t;
  if (tg < 192) { s = 0; t = tg; }
  else if (tg < 576) { s = 1; t = tg - 192; }
  else { s = 2; t = tg - 576; }
  int T = 192 << s;
  int k = (kk < 32) ? kk : kk - 32;
  int r = (k * t) % T;  // exact angle reduction
  float th = TWO_PI * (float)r / (float)T;
  Bas[gid] = (kk < 32) ? cosf(th) : -sinf(th);
}

// Weff[s][i*64+kk][o]; kk<32 -> alpha*(c*wr - sn*wi); kk>=32 -> -alpha*(c*wi + sn*wr)
__global__ void k_weff(const float* __restrict__ wr_, const float* __restrict__ wi_,
                       float* __restrict__ Weff) {
  int gid = blockIdx.x * 256 + threadIdx.x;  // 3*256*64*256
  int o = gid & 255;
  int rest = gid >> 8;
  int kk = rest & 63;
  rest >>= 6;
  int i = rest & 255;
  int s = rest >> 8;
  int T = 192 << s;
  int k = (kk < 32) ? kk : kk - 32;
  int r = (k * (OUTLEN - 1)) % T;
  float th = TWO_PI * (float)r / (float)T;
  float c = cosf(th), sn = sinf(th);
  float al = ((k == 0) ? 1.0f : 2.0f) / (float)T;
  size_t wofs = (((size_t)s * 256 + i) * 256 + o) * 32 + k;
  float wr = wr_[wofs], wi = wi_[wofs];
  float val = (kk < 32) ? al * (c * wr - sn * wi) : -al * (c * wi + sn * wr);
  Weff[((size_t)s * 16384 + i * 64 + kk) * 256 + o] = val;
}

// ---------------------------------------------------------------------------
// Stage 1: HiPPO scan. 96 WGs = 3 scales x 32 channel-groups, 256 thr (8 waves)
// LDS: sAdTs = AdT pre-swizzled into per-lane B fragments (one b64 per operand,
//      conflict-free); sCA = state in A-fragment row-major layout (stride 260
//      == 4 mod 64 -> the wave's 32 b64 reads cover all 64 banks once).
// cs slice layout: [grp(32)][n(256)][m(16)]  (flat 131072 per time step)
// ---------------------------------------------------------------------------
__global__ __launch_bounds__(256) void k_scan(const float* __restrict__ inp,
                                              const float* __restrict__ AdT_all,
                                              const float* __restrict__ Bd_all,
                                              float* __restrict__ cs_all) {
  __shared__ float sAdTs[HID * HID];  // 256 KB, pair idx ((kt*16+ntg)*32+lane)
  __shared__ float sCA[16][260];      // state C[m][k], A-fragment friendly
  __shared__ float sBd[HID];
  __shared__ float sF[16];

  const int tid = threadIdx.x;
  const int s = blockIdx.x >> 5;
  const int grp = blockIdx.x & 31;
  const int T = 192 << s;
  const float* AdT = AdT_all + (size_t)s * HID * HID;
  float* cs = cs_all + (size_t)tprefix(s) * TSLICE;
  const int ch0 = grp * 16;
  const int b = ch0 >> 6, j0 = ch0 & 63;

  // swizzle AdT into per-lane B-fragment pairs
  for (int p = tid; p < 32768; p += 256) {
    int lane_ = p & 31;
    int ntg = (p >> 5) & 15;
    int kt = p >> 9;
    int n = ntg * 16 + (lane_ & 15);
    int kb = kt * 4 + 2 * (lane_ >> 4);
    sAdTs[2 * p] = AdT[(size_t)kb * HID + n];
    sAdTs[2 * p + 1] = AdT[(size_t)(kb + 1) * HID + n];
  }
  sBd[tid] = Bd_all[s * HID + tid];
  for (int i = tid; i < 16 * 260; i += 256) (&sCA[0][0])[i] = 0.0f;
  __syncthreads();

  const int wave = tid >> 5, lane = tid & 31;
  const int lh = lane >> 4, lm = lane & 15;
  const int n0 = wave * 32, n1 = n0 + 16;
  const int nt0 = wave * 2, nt1 = nt0 + 1;
  const v2f* Bs = (const v2f*)sAdTs;

  for (int t = 0; t < T; ++t) {
    if (tid < 16)
      sF[tid] = inp[((size_t)b * INLEN + (INLEN - T + t)) * 64 + j0 + tid];

    v8f a0a = {}, a0b = {}, a1a = {}, a1b = {};  // 4 independent chains
#pragma unroll 4
    for (int kt = 0; kt < 64; kt += 2) {
      v2f aA = *(const v2f*)&sCA[lm][kt * 4 + 2 * lh];
      v2f aB = *(const v2f*)&sCA[lm][kt * 4 + 4 + 2 * lh];
      v2f b0A = Bs[(kt * 16 + nt0) * 32 + lane];
      v2f b1A = Bs[(kt * 16 + nt1) * 32 + lane];
      v2f b0B = Bs[(kt * 16 + 16 + nt0) * 32 + lane];
      v2f b1B = Bs[(kt * 16 + 16 + nt1) * 32 + lane];
      a0a = wmma4(aA, b0A, a0a);
      a1a = wmma4(aA, b1A, a1a);
      a0b = wmma4(aB, b0B, a0b);
      a1b = wmma4(aB, b1B, a1b);
    }
    __syncthreads();  // all reads of old state done; sF visible

    float r0[8], r1[8];
#pragma unroll
    for (int v = 0; v < 8; ++v) {
      int m = v + 8 * lh;
      float fb = sF[m];
      float o0 = a0a[v] + a0b[v] + fb * sBd[n0 + lm];
      float o1 = a1a[v] + a1b[v] + fb * sBd[n1 + lm];
      sCA[m][n0 + lm] = o0;
      sCA[m][n1 + lm] = o1;
      r0[v] = o0;
      r1[v] = o1;
    }
    // group-transposed slice: contiguous 8 floats per lane -> b128 stores
    float* csb = cs + (size_t)t * TSLICE + (size_t)grp * 4096;
    v4f q0 = {r0[0], r0[1], r0[2], r0[3]};
    v4f q1 = {r0[4], r0[5], r0[6], r0[7]};
    v4f q2 = {r1[0], r1[1], r1[2], r1[3]};
    v4f q3 = {r1[4], r1[5], r1[6], r1[7]};
    *(v4f*)(csb + (n0 + lm) * 16 + 8 * lh) = q0;
    *(v4f*)(csb + (n0 + lm) * 16 + 8 * lh + 4) = q1;
    *(v4f*)(csb + (n1 + lm) * 16 + 8 * lh) = q2;
    *(v4f*)(csb + (n1 + lm) * 16 + 8 * lh + 4) = q3;
    __syncthreads();  // new state visible before next step
  }
}

// ---------------------------------------------------------------------------
// Stage 2: XF[m'][kk] = sum_t cs[t][m'] * Bas[t][kk]   (m' = permuted batch row)
// 3072 WGs (1024/scale), 8 waves each own a 16-row M tile across all 64 cols.
// No LDS on purpose: HBM-bound (re-reads cs), keep occupancy high.
// ---------------------------------------------------------------------------
__global__ __launch_bounds__(256) void k_dft(const float* __restrict__ cs_all,
                                             const float* __restrict__ Bas_all,
                                             float* __restrict__ XF) {
  const int blk = blockIdx.x;
  const int s = blk >> 10;
  const int bs = blk & 1023;
  const int T = 192 << s;
  const float* cs = cs_all + (size_t)tprefix(s) * TSLICE;
  const float* Bas = Bas_all + (size_t)tprefix(s) * 64;
  const int tid = threadIdx.x, wave = tid >> 5, lane = tid & 31;
  const int lh = lane >> 4, lm = lane & 15;
  const int m0 = (bs * 8 + wave) * 16;  // < 131072

  v8f acc[4] = {};
  for (int kt = 0; kt < T / 4; ++kt) {
    int kb = kt * 4 + 2 * lh;
    v2f a;
    a.x = cs[(size_t)kb * TSLICE + m0 + lm];
    a.y = cs[(size_t)(kb + 1) * TSLICE + m0 + lm];
#pragma unroll
    for (int nt = 0; nt < 4; ++nt) {
      v2f bb = {Bas[kb * 64 + nt * 16 + lm], Bas[(kb + 1) * 64 + nt * 16 + lm]};
      acc[nt] = wmma4(a, bb, acc[nt]);
    }
  }
  float* xf = XF + (size_t)s * 131072 * 64;
#pragma unroll
  for (int nt = 0; nt < 4; ++nt)
#pragma unroll
    for (int v = 0; v < 8; ++v)
      xf[(size_t)(m0 + v + 8 * lh) * 64 + nt * 16 + lm] = acc[nt][v];
}

// ---------------------------------------------------------------------------
// Stage 3: xout[ch][o] = sum_{i,kk} XF[row(ch,i)][kk] * Weff[i*64+kk][o]
// row(ch,i) = (ch>>4)*4096 + i*16 + (ch&15)   (matches scan's cs permutation)
// 192 WGs (64/scale), one 16x16 tile per wave, 4 accumulator chains.
// ---------------------------------------------------------------------------
__global__ __launch_bounds__(256) void k_spec(const float* __restrict__ XF,
                                              const float* __restrict__ Weff,
                                              float* __restrict__ xout) {
  const int blk = blockIdx.x;
  const int s = blk >> 6;
  const int bs = blk & 63;
  const int tid = threadIdx.x, wave = tid >> 5, lane = tid & 31;
  const int lh = lane >> 4, lm = lane & 15;
  const int tileId = bs * 8 + wave;  // 0..511
  const int mt = tileId >> 4;        // ch tile == grp
  const int o0 = (tileId & 15) * 16;
  const float* xf = XF + (size_t)s * 131072 * 64;
  const float* wv = Weff + (size_t)s * 16384 * 256;
  const size_t abase = (size_t)mt * 262144 + (size_t)lm * 64;  // (mt*4096+lm)*64

  v8f acc[4] = {};
  for (int K0 = 0; K0 < 16384; K0 += 16) {
#pragma unroll
    for (int j = 0; j < 4; ++j) {
      int kb = K0 + 4 * j + 2 * lh;  // even; K and K+1 share the same i-block
      size_t aoff = abase + (size_t)(kb >> 6) * 1024 + (kb & 63);
      v2f a = {xf[aoff], xf[aoff + 1]};
      v2f bb = {wv[(size_t)kb * 256 + o0 + lm],
                wv[(size_t)(kb + 1) * 256 + o0 + lm]};
      acc[j] = wmma4(a, bb, acc[j]);
    }
  }
  float* xo = xout + (size_t)s * NCH * HID;
#pragma unroll
  for (int v = 0; v < 8; ++v)
    xo[(size_t)(mt * 16 + v + 8 * lh) * HID + o0 + lm] =
        (acc[0][v] + acc[1][v]) + (acc[2][v] + acc[3][v]);
}

// ---------------------------------------------------------------------------
// Stage 4: out[b][t][j] = sum_s mlp_w[s] * (xout_s[ch] . em_s[t]) + mlp_b
// 48 WGs, 384 wave-tiles (32 ch-tiles x 12 t-tiles), 2 chains per scale.
// ---------------------------------------------------------------------------
__global__ __launch_bounds__(256) void k_final(const float* __restrict__ xout,
                                               const float* __restrict__ em,
                                               const float* __restrict__ mlw,
                                               const float* __restrict__ mlb,
                                               float* __restrict__ out) {
  const int tid = threadIdx.x, wave = tid >> 5, lane = tid & 31;
  const int lh = lane >> 4, lm = lane & 15;
  const int tileId = blockIdx.x * 8 + wave;  // 0..383
  const int m0 = (tileId / 12) * 16;
  const int t0 = (tileId % 12) * 16;

  float res[8] = {0.f, 0.f, 0.f, 0.f, 0.f, 0.f, 0.f, 0.f};
  for (int s = 0; s < 3; ++s) {
    const float* xo = xout + (size_t)s * NCH * HID;
    const float* ems = em + (size_t)s * OUTLEN * 256;
    v8f acc0 = {}, acc1 = {};
    for (int k = 0; k < 256; k += 8) {
      int kb0 = k + 2 * lh;
      int kb1 = kb0 + 4;
      v2f a0 = {xo[(size_t)(m0 + lm) * HID + kb0],
                xo[(size_t)(m0 + lm) * HID + kb0 + 1]};
      v2f a1 = {xo[(size_t)(m0 + lm) * HID + kb1],
                xo[(size_t)(m0 + lm) * HID + kb1 + 1]};
      v2f b0 = {ems[(size_t)(t0 + lm) * 256 + kb0],
                ems[(size_t)(t0 + lm) * 256 + kb0 + 1]};
      v2f b1 = {ems[(size_t)(t0 + lm) * 256 + kb1],
                ems[(size_t)(t0 + lm) * 256 + kb1 + 1]};
      acc0 = wmma4(a0, b0, acc0);
      acc1 = wmma4(a1, b1, acc1);
    }
    float w = mlw[s];
#pragma unroll
    for (int v = 0; v < 8; ++v) res[v] += w * (acc0[v] + acc1[v]);
  }
  float bias = mlb[0];
#pragma unroll
  for (int v = 0; v < 8; ++v) {
    int ch = m0 + v + 8 * lh;
    int t = t0 + lm;
    out[((size_t)(ch >> 6) * OUTLEN + t) * 64 + (ch & 63)] = res[v] + bias;
  }
}

// ---------------------------------------------------------------------------
extern "C" void kernel_launch(void* const* d_in, const int* in_sizes, int n_in,
                              void* d_out, int out_size, void* d_ws, size_t ws_size,
                              hipStream_t stream) {
  (void)in_sizes; (void)n_in; (void)out_size; (void)ws_size;
  const float* inputs = (const float*)d_in[0];
  const float* spec_wr = (const float*)d_in[1];
  const float* spec_wi = (const float*)d_in[2];
  const float* mlp_w = (const float*)d_in[3];
  const float* mlp_b = (const float*)d_in[4];
  float* out = (float*)d_out;

  char* base = (char*)d_ws;
  size_t off = 0;
  auto carve = [&](size_t bytes) -> char* {
    char* p = base + off;
    off = (off + bytes + 255) & ~(size_t)255;
    return p;
  };
  double* G    = (double*)carve((size_t)3 * 256 * 513 * sizeof(double));
  float* AdT   = (float*)carve((size_t)3 * 65536 * sizeof(float));
  float* Bd    = (float*)carve((size_t)3 * 256 * sizeof(float));
  float* em    = (float*)carve((size_t)3 * OUTLEN * 256 * sizeof(float));
  float* Bas   = (float*)carve((size_t)1344 * 64 * sizeof(float));
  float* Weff  = (float*)carve((size_t)3 * 16384 * 256 * sizeof(float));
  float* XF    = (float*)carve((size_t)3 * 131072 * 64 * sizeof(float));
  float* xoutb = (float*)carve((size_t)3 * NCH * HID * sizeof(float));
  float* cs    = (float*)carve((size_t)1344 * TSLICE * sizeof(float));

  k_gauss<<<3, 256, 0, stream>>>(G, AdT, Bd);
  k_em<<<3, OUTLEN, 0, stream>>>(em);
  k_bas<<<(1344 * 64) / 256, 256, 0, stream>>>(Bas);
  k_weff<<<(3 * 256 * 64 * 256) / 256, 256, 0, stream>>>(spec_wr, spec_wi, Weff);
  k_scan<<<96, 256, 0, stream>>>(inputs, AdT, Bd, cs);
  k_dft<<<3072, 256, 0, stream>>>(cs, Bas, XF);
  k_spec<<<192, 256, 0, stream>>>(XF, Weff, xoutb);
  k_final<<<48, 256, 0, stream>>>(xoutb, em, mlp_w, mlp_b, out);
}